// Polynomial_module_54425825575575
// MI455X (gfx1250) — compile-verified
//
#include <hip/hip_runtime.h>
#include <hip/hip_bf16.h>
#include <math.h>

// ---------------------------------------------------------------------------
// Problem (from reference): N=8192 pts, K=16 NN, sigma2 = 1.5, scalar f32 out.
// All-pairs NN via V_WMMA_F32_16X16X4_F32; spare K-slots of the K=4 MAC carry
// ||ch2m||^2 so the WMMA emits the selection key directly:
//   key[m][j] = ||ch2m_m||^2 - 2 * dot(ch2m_m, ch1_j)
// A-operand chunks are staged into LDS with GLOBAL_LOAD_ASYNC_TO_LDS_B128
// (ASYNCcnt double-buffering), shared by 8 waves per workgroup; the per-tile
// LDS fetch is software-pipelined one iteration ahead of the WMMA.
// ---------------------------------------------------------------------------

typedef __attribute__((ext_vector_type(2))) float v2f;
typedef __attribute__((ext_vector_type(8))) float v8f;

#define KNN_K 16
#define SIGMA2 1.5f

#define WAVES_PER_BLOCK 8
#define BLOCK_THREADS   (WAVES_PER_BLOCK * 32)
#define CHUNK_TILES     64                          // tiles of packedA per chunk
#define CHUNK_BYTES     (CHUNK_TILES * 32 * 8)      // 16 KB
#define ASYNC_PER_CHUNK (CHUNK_BYTES / (BLOCK_THREADS * 16))  // 4 b128 per thread

// ---------------------------------------------------------------------------
// Kernel A: bilinear polynomial + WMMA-ready packed A operand.
// packedA tile t (32 v2f, one per lane):
//   lanes  0..15 : (-2*y1, -2*y2)  -> A 16x4 K=0 (VGPR0) / K=1 (VGPR1)
//   lanes 16..31 : (||y||^2, 0)    -> A 16x4 K=2 (VGPR0) / K=3 (VGPR1)
// ---------------------------------------------------------------------------
__global__ __launch_bounds__(256) void poly_pack_kernel(
    const float* __restrict__ ch2, const float* __restrict__ M1,
    const float* __restrict__ M2, float* __restrict__ ch2m,
    v2f* __restrict__ packedA, int N) {
  int m = blockIdx.x * blockDim.x + threadIdx.x;
  if (m >= N) return;
  float x1 = ch2[2 * m + 0];
  float x2 = ch2[2 * m + 1];
  float y1 = M1[0] + M1[1] * x2 + M1[2] * x1 + M1[3] * x1 * x2;
  float y2 = M2[0] + M2[1] * x2 + M2[2] * x1 + M2[3] * x1 * x2;
  ch2m[2 * m + 0] = y1;
  ch2m[2 * m + 1] = y2;
  int t = m >> 4;
  int r = m & 15;
  v2f lo;  lo.x = -2.0f * y1;        lo.y = -2.0f * y2;
  v2f hi;  hi.x = y1 * y1 + y2 * y2; hi.y = 0.0f;
  packedA[t * 32 + r]      = lo;
  packedA[t * 32 + 16 + r] = hi;
}

// Issue one 16 KB chunk of packedA into LDS: 4 coalesced b128 async loads
// per thread, tracked with ASYNCcnt.
static __device__ __forceinline__ void issue_chunk_async(
    const char* gBase, unsigned ldsBase, int tid) {
#pragma unroll
  for (int r = 0; r < ASYNC_PER_CHUNK; ++r) {
    unsigned off = (unsigned)(r * BLOCK_THREADS * 16 + tid * 16);
    unsigned ldsAddr = ldsBase + off;
    const char* gAddr = gBase + off;
    asm volatile("global_load_async_to_lds_b128 %0, %1, off"
                 :: "v"(ldsAddr), "v"(gAddr) : "memory");
  }
}

// ---------------------------------------------------------------------------
// Kernel B: 8 waves per block; wave w owns ch1 tile blockIdx.x*8+w.
// All waves share double-buffered LDS staging of packedA chunks.
// Per V_WMMA_F32_16X16X4_F32: lane l accumulates 8 keys for ch1 point l&15
// (rows v + 8*(l>=16)); per-lane sorted top-16; lanes l/l+16 merged via LDS.
// ---------------------------------------------------------------------------
__global__ __launch_bounds__(BLOCK_THREADS) void knn_wmma_kernel(
    const float* __restrict__ ch1, const v2f* __restrict__ packedA,
    int* __restrict__ idxout, int nTiles) {
  __shared__ __align__(16) v2f stage[2][CHUNK_TILES * 32];   // 2 x 16 KB

  const int tid  = threadIdx.x;
  const int lane = tid & 31;
  const int wv   = tid >> 5;
  const int col  = lane & 15;
  const bool hi  = lane >= 16;
  const int tileB = blockIdx.x * WAVES_PER_BLOCK + wv;

  // B operand (fixed), 4x16 f32:
  //   lanes 0-15 : (x_j, y_j)  -> K=0/K=1 ; lanes 16-31: (1,0) -> K=2/K=3
  v2f b;
  if (!hi) {
    b.x = ch1[(tileB * 16 + col) * 2 + 0];
    b.y = ch1[(tileB * 16 + col) * 2 + 1];
  } else {
    b.x = 1.0f;
    b.y = 0.0f;
  }

  float dist[KNN_K];
  int   ind[KNN_K];
#pragma unroll
  for (int j = 0; j < KNN_K; ++j) { dist[j] = 3.0e38f; ind[j] = 0x7fffffff; }

  const int nChunks = nTiles / CHUNK_TILES;   // 512/64 = 8 (N=8192)
  const char* gsrc = (const char*)packedA;
  const unsigned lds0 = (unsigned)(uintptr_t)&stage[0][0];
  const unsigned lds1 = (unsigned)(uintptr_t)&stage[1][0];

  // Prologue: chunk 0 in flight.
  issue_chunk_async(gsrc, lds0, tid);

  for (int c = 0; c < nChunks; ++c) {
    __syncthreads();   // all waves done reading buffer we are about to refill
    if (c + 1 < nChunks) {
      issue_chunk_async(gsrc + (size_t)(c + 1) * CHUNK_BYTES,
                        ((c + 1) & 1) ? lds1 : lds0, tid);
      // async loads complete in order: <=4 outstanding => chunk c landed.
      asm volatile("s_wait_asynccnt 4" ::: "memory");
    } else {
      asm volatile("s_wait_asynccnt 0" ::: "memory");
    }
    __syncthreads();   // every wave's portion of chunk c visible in LDS

    const v2f* buf = stage[c & 1];
    const int tBase = c * CHUNK_TILES;

    // Software-pipelined tile fetch: a holds tile tl, preload tl+1 before the
    // WMMA so the ds_load overlaps WMMA + insertion (wrap preload is a dummy).
    v2f a = buf[lane];
#pragma unroll 2
    for (int tl = 0; tl < CHUNK_TILES; ++tl) {
      v2f a_next = buf[((tl + 1) & (CHUNK_TILES - 1)) * 32 + lane];
      v8f cacc = {};
      v8f d = __builtin_amdgcn_wmma_f32_16x16x4_f32(
          /*neg_a=*/false, a, /*neg_b=*/false, b,
          /*c_mod=*/(short)0, cacc, /*reuse_a=*/false, /*reuse_b=*/false);

      const int rowBase = (tBase + tl) * 16 + (hi ? 8 : 0);
#pragma unroll
      for (int v = 0; v < 8; ++v) {
        float key = d[v];
        int   cand = rowBase + v;
        if (key < dist[KNN_K - 1]) {
          // branch-free sorted insert; strict < keeps earlier (smaller)
          // indices ahead on ties, matching top_k stability.
#pragma unroll
          for (int j = KNN_K - 1; j >= 1; --j) {
            bool lt_j   = key < dist[j];
            bool lt_jm1 = key < dist[j - 1];
            float nd = lt_jm1 ? dist[j - 1] : key;
            int   ni = lt_jm1 ? ind[j - 1] : cand;
            dist[j] = lt_j ? nd : dist[j];
            ind[j]  = lt_j ? ni : ind[j];
          }
          bool lt0 = key < dist[0];
          dist[0] = lt0 ? key : dist[0];
          ind[0]  = lt0 ? cand : ind[0];
        }
      }
      a = a_next;
    }
  }

  // Reuse the 32 KB staging LDS as per-wave merge scratch (4 KB per wave).
  __syncthreads();
  float* sdist = (float*)((char*)&stage[0][0] + wv * 4096);
  int*   sind  = (int*)((char*)&stage[0][0] + wv * 4096 + 2048);
#pragma unroll
  for (int j = 0; j < KNN_K; ++j) {
    sdist[lane * KNN_K + j] = dist[j];
    sind[lane * KNN_K + j]  = ind[j];
  }
  __syncthreads();

  if (lane < 16) {
    int i = 0, j2 = 0;
    const int outBase = (tileB * 16 + lane) * KNN_K;
#pragma unroll
    for (int k = 0; k < KNN_K; ++k) {
      float da = sdist[lane * KNN_K + i];
      float db = sdist[(lane + 16) * KNN_K + j2];
      int   ia = sind[lane * KNN_K + i];
      int   ib = sind[(lane + 16) * KNN_K + j2];
      bool takeA = (da < db) || (da == db && ia < ib);
      idxout[outBase + k] = takeA ? ia : ib;
      i  += takeA ? 1 : 0;
      j2 += takeA ? 0 : 1;
    }
  }
}

// ---------------------------------------------------------------------------
// Kernel C: reference's scrambled gather, reproduced exactly:
// knn[k,n] uses pair (idxflat[k*N + n], n)  [idx.reshape(-1) is flat RM].
// ---------------------------------------------------------------------------
__global__ __launch_bounds__(256) void kl_kernel(
    const float* __restrict__ ch1, const float* __restrict__ ch2m,
    const int* __restrict__ idxflat, float* __restrict__ partial, int N) {
  __shared__ float red[256];
  int n = blockIdx.x * 256 + threadIdx.x;
  float val = 0.0f;
  if (n < N) {
    float x = ch1[2 * n + 0];
    float y = ch1[2 * n + 1];
    const float scale = 0.5f / (SIGMA2 * SIGMA2);
    float s = 0.0f;
#pragma unroll
    for (int k = 0; k < KNN_K; ++k) {
      int a = idxflat[k * N + n];
      float dx = x - ch2m[2 * a + 0];
      float dy = y - ch2m[2 * a + 1];
      float D = (dx * dx + dy * dy) * scale;
      s += expf(-D);
    }
    float expD = s / (float)N;
    val = (expD != 0.0f) ? -logf(expD) : 0.0f;
  }
  red[threadIdx.x] = val;
  __syncthreads();
#pragma unroll
  for (int s = 128; s > 0; s >>= 1) {
    if (threadIdx.x < s) red[threadIdx.x] += red[threadIdx.x + s];
    __syncthreads();
  }
  if (threadIdx.x == 0) partial[blockIdx.x] = red[0];
}

// Deterministic fixed-order final reduction (no float atomics).
__global__ void final_reduce_kernel(const float* __restrict__ partial,
                                    float* __restrict__ out, int nb) {
  if (threadIdx.x == 0 && blockIdx.x == 0) {
    float s = 0.0f;
    for (int i = 0; i < nb; ++i) s += partial[i];
    out[0] = s;
  }
}

// ---------------------------------------------------------------------------
extern "C" void kernel_launch(void* const* d_in, const int* in_sizes, int n_in,
                              void* d_out, int out_size, void* d_ws, size_t ws_size,
                              hipStream_t stream) {
  const float* ch1 = (const float*)d_in[0];
  const float* ch2 = (const float*)d_in[1];
  const float* M1  = (const float*)d_in[2];
  const float* M2  = (const float*)d_in[3];
  const int N = in_sizes[0] / 2;       // 8192
  const int nTiles = N / 16;           // 512

  // Workspace layout (~704 KB, L2-resident)
  char* ws = (char*)d_ws;
  float* ch2m    = (float*)ws;                                  // 2N floats
  v2f*   packedA = (v2f*)(ws + (size_t)2 * N * sizeof(float));  // 2N v2f
  int*   idxflat = (int*)((char*)packedA + (size_t)2 * N * sizeof(v2f));
  float* partial = (float*)((char*)idxflat + (size_t)N * KNN_K * sizeof(int));

  const int nBlocksPts = (N + 255) / 256;
  const int nBlocksKnn = nTiles / WAVES_PER_BLOCK;  // 64

  poly_pack_kernel<<<nBlocksPts, 256, 0, stream>>>(ch2, M1, M2, ch2m, packedA, N);
  knn_wmma_kernel<<<nBlocksKnn, BLOCK_THREADS, 0, stream>>>(ch1, packedA, idxflat, nTiles);
  kl_kernel<<<nBlocksPts, 256, 0, stream>>>(ch1, ch2m, idxflat, partial, N);
  final_reduce_kernel<<<1, 32, 0, stream>>>(partial, (float*)d_out, nBlocksPts);
}